// _PixelCNNDistribution_51634096833163
// MI455X (gfx1250) — compile-verified
//
#include <hip/hip_runtime.h>
#include <math.h>

#define NMIX 10
#define BB 32
#define CH 3
#define HH 128
#define WW 128
#define HW (HH * WW)             // 16384
#define NPIX (BB * HW)           // 524288
#define THREADS 256
#define NBLOCKS (NPIX / THREADS) // 2048

typedef float v2f __attribute__((ext_vector_type(2)));
typedef float v8f __attribute__((ext_vector_type(8)));

// ---- fast device math (branchless, native trans ops) ----

__device__ __forceinline__ float fast_rcp(float x) {
#if __has_builtin(__builtin_amdgcn_rcpf)
    return __builtin_amdgcn_rcpf(x);   // v_rcp_f32
#else
    return 1.0f / x;
#endif
}

__device__ __forceinline__ float fast_tanh(float x) {
#if __has_builtin(__builtin_amdgcn_tanhf)
    return __builtin_amdgcn_tanhf(x);  // v_tanh_f32 (CDNA5 trans op)
#else
    // tanh(x) = sign(x) * (1-t)/(1+t), t = exp(-2|x|)
    const float t = __expf(-2.0f * fabsf(x));
    return copysignf((1.0f - t) * fast_rcp(1.0f + t), x);
#endif
}

// sigmoid(y) and softplus(y) sharing a single exp:
//   t = exp(-|y|)
//   softplus(y) = max(y,0) + log(1+t)
//   sigmoid(y)  = (y>=0) ? 1/(1+t) : 1 - 1/(1+t)
__device__ __forceinline__ void sig_sp(float y, float& sig, float& sp) {
    const float t = __expf(-fabsf(y));
    const float u = 1.0f + t;
    const float r = fast_rcp(u);
    sig = (y >= 0.0f) ? r : (1.0f - r);
    sp  = fmaxf(y, 0.0f) + __logf(u);
}

__device__ __forceinline__ float sp_only(float y) {
    const float t = __expf(-fabsf(y));
    return fmaxf(y, 0.0f) + __logf(1.0f + t);
}

// Kernel 1: one thread per pixel. Streams 100 param channels + 3 sample
// channels (all lane-coalesced, channel stride = HW), computes the
// discretized-logistic-mixture log-prob, reduces to one partial per block.
__global__ __launch_bounds__(THREADS)
void dmll_main(const float* __restrict__ samples,
               const float* __restrict__ params,
               float* __restrict__ partials) {
    const int idx = blockIdx.x * THREADS + threadIdx.x;
    const int b   = idx >> 14;          // / HW
    const int rem = idx & (HW - 1);     // h*W + w

    const float* sp = samples + (size_t)b * CH * HW + rem;
    const float* pp = params  + (size_t)b * (10 * NMIX) * HW + rem;

    // Warm L2 for the mid/late channels while early mixtures compute.
    __builtin_prefetch(pp + (size_t)40 * HW, 0, 0);
    __builtin_prefetch(pp + (size_t)80 * HW, 0, 0);

    const float x0 = sp[0]              * 2.0f - 1.0f;
    const float x1 = sp[(size_t)HW]     * 2.0f - 1.0f;
    const float x2 = sp[(size_t)2 * HW] * 2.0f - 1.0f;
    const float xs[3] = {x0, x1, x2};

    float lp[NMIX];
    float t[NMIX];
#pragma unroll
    for (int m = 0; m < NMIX; ++m) lp[m] = pp[(size_t)m * HW];

    const float LOG127_5 = 4.852030263919617f; // log(128)

#pragma unroll
    for (int m = 0; m < NMIX; ++m) {
        const float mean0 = pp[(size_t)(10 + 0 * 30 +      m) * HW];
        const float ls0   = fmaxf(pp[(size_t)(10 + 0 * 30 + 10 + m) * HW], -7.0f);
        const float c0    = fast_tanh(pp[(size_t)(10 + 0 * 30 + 20 + m) * HW]);
        const float mean1 = pp[(size_t)(10 + 1 * 30 +      m) * HW];
        const float ls1   = fmaxf(pp[(size_t)(10 + 1 * 30 + 10 + m) * HW], -7.0f);
        const float c1    = fast_tanh(pp[(size_t)(10 + 1 * 30 + 20 + m) * HW]);
        const float mean2 = pp[(size_t)(10 + 2 * 30 +      m) * HW];
        const float ls2   = fmaxf(pp[(size_t)(10 + 2 * 30 + 10 + m) * HW], -7.0f);
        const float c2    = fast_tanh(pp[(size_t)(10 + 2 * 30 + 20 + m) * HW]);

        float means[3];
        means[0] = mean0;
        means[1] = mean1 + c0 * x0;
        means[2] = mean2 + c1 * x0 + c2 * x1;
        const float lss[3] = {ls0, ls1, ls2};

        float chsum = 0.0f;
#pragma unroll
        for (int ch = 0; ch < 3; ++ch) {
            const float x        = xs[ch];
            const float centered = x - means[ch];
            const float inv      = __expf(-lss[ch]);
            const float plus_in  = inv * (centered + 1.0f / 255.0f);
            const float min_in   = inv * (centered - 1.0f / 255.0f);
            const float mid_in   = inv * centered;

            float cdf_plus, sp_plus, cdf_min, sp_min;
            sig_sp(plus_in, cdf_plus, sp_plus);
            sig_sp(min_in,  cdf_min,  sp_min);

            const float cdf_delta      = cdf_plus - cdf_min;
            const float log_cdf_plus   = plus_in - sp_plus;
            const float log_om_cdf_min = -sp_min;
            const float log_pdf_mid    = mid_in - lss[ch] - 2.0f * sp_only(mid_in);

            const float inner = (cdf_delta > 1e-5f)
                                    ? __logf(fmaxf(cdf_delta, 1e-12f))
                                    : (log_pdf_mid - LOG127_5);
            const float v = (x < -0.999f) ? log_cdf_plus
                          : ((x > 0.999f) ? log_om_cdf_min : inner);
            chsum += v;
        }
        t[m] = chsum + lp[m];
    }

    // pixel log-prob = LSE(t) - LSE(lp)   (log_softmax folded into LSE)
    float mt = t[0], ml = lp[0];
#pragma unroll
    for (int m = 1; m < NMIX; ++m) {
        mt = fmaxf(mt, t[m]);
        ml = fmaxf(ml, lp[m]);
    }
    float st = 0.0f, sl = 0.0f;
#pragma unroll
    for (int m = 0; m < NMIX; ++m) {
        st += __expf(t[m] - mt);
        sl += __expf(lp[m] - ml);
    }
    float val = (mt + __logf(st)) - (ml + __logf(sl));

    // wave32 reduction (deterministic fixed order)
#pragma unroll
    for (int off = 16; off > 0; off >>= 1)
        val += __shfl_down(val, off, 32);

    __shared__ float wsum[THREADS / 32];
    const int lane = threadIdx.x & 31;
    const int wid  = threadIdx.x >> 5;
    if (lane == 0) wsum[wid] = val;
    __syncthreads();
    if (threadIdx.x == 0) {
        float s = 0.0f;
#pragma unroll
        for (int w = 0; w < THREADS / 32; ++w) s += wsum[w];
        partials[blockIdx.x] = s;
    }
}

// Kernel 2: single wave32. Reduce NBLOCKS partials deterministically using
// V_WMMA_F32_16X16X4_F32: C += A(16x4 partials) x B(4x16 ones). 64 values per
// WMMA. D[M][N] = sum of row M over all chunks (every column identical), so
// total = sum_M D[M][0] = (lane0 rows 0-7) + (lane16 rows 8-15).
__global__ __launch_bounds__(32)
void dmll_reduce(const float* __restrict__ partials, float* __restrict__ out) {
    const int lane = threadIdx.x;

    v2f ones;
    ones.x = 1.0f;
    ones.y = 1.0f;
    v8f acc = {};

    for (int base = 0; base < NBLOCKS; base += 64) {
        v2f a;
        a.x = partials[base + lane];
        a.y = partials[base + 32 + lane];
        acc = __builtin_amdgcn_wmma_f32_16x16x4_f32(
            /*neg_a=*/false, a, /*neg_b=*/false, ones,
            /*c_mod=*/(short)0, acc, /*reuse_a=*/false, /*reuse_b=*/false);
    }

    float s = acc[0] + acc[1] + acc[2] + acc[3] +
              acc[4] + acc[5] + acc[6] + acc[7];
    const float other = __shfl(s, 16, 32); // lane 0 picks up lane 16 (rows 8-15)
    if (lane == 0) out[0] = s + other;
}

extern "C" void kernel_launch(void* const* d_in, const int* in_sizes, int n_in,
                              void* d_out, int out_size, void* d_ws, size_t ws_size,
                              hipStream_t stream) {
    const float* samples = (const float*)d_in[0]; // (32, 3, 128, 128) f32
    const float* params  = (const float*)d_in[1]; // (32, 100, 128, 128) f32
    float* out      = (float*)d_out;              // scalar f32
    float* partials = (float*)d_ws;               // NBLOCKS floats (8 KB)

    dmll_main<<<NBLOCKS, THREADS, 0, stream>>>(samples, params, partials);
    dmll_reduce<<<1, 32, 0, stream>>>(partials, out);
}